// KerRW_37349035606680
// MI455X (gfx1250) — compile-verified
//
#include <hip/hip_runtime.h>

typedef __attribute__((ext_vector_type(2))) float v2f;
typedef __attribute__((ext_vector_type(4))) float v4f;
typedef __attribute__((ext_vector_type(8))) float v8f;
typedef __attribute__((ext_vector_type(4))) unsigned int v4u;
typedef __attribute__((ext_vector_type(4))) int v4i;
typedef __attribute__((ext_vector_type(8))) int v8i;

#define HG 16
#define SHG 10
#define HD 64
#define NSTEP 3
#define NMAT (NSTEP * HG) /* 48 */
#define NTRI (SHG * (SHG - 1) / 2) /* 45 */

#if defined(__has_builtin)
#if __has_builtin(__builtin_amdgcn_tensor_load_to_lds)
#define HAVE_TDM 1
#endif
#endif

// ---------------------------------------------------------------------------
// Phase 1: per hidden graph g, build A (relu'd symmetric 10x10), iterate
// Z_{k+1} = A Z_k, and emit M[g,step] = Z0^T Z_{step+1} (64x64) directly in
// WMMA-B-operand swizzled order into d_ws:
//   ws[km*4096 + (k4*4 + t)*64 + lane*2 + j] = M[4*k4 + 2*(lane>>4) + j][16*t + (lane&15)]
// so phase 2's B fetches are contiguous per-lane ds_load_b64 (bank-conflict free).
// ---------------------------------------------------------------------------
__global__ __launch_bounds__(256) void rw_prep(
    const float* __restrict__ adj_hidden,
    const float* __restrict__ features_hidden,
    float* __restrict__ msw) {
  __shared__ float A[SHG * SHG];
  __shared__ float Z0[SHG * HD];
  __shared__ float Zb[2][SHG * HD];
  const int tid = threadIdx.x;
  const int g = blockIdx.x;

  if (tid < SHG * SHG) A[tid] = 0.f;
  for (int i = tid; i < SHG * HD; i += 256) {
    float z = features_hidden[(size_t)g * SHG * HD + i];
    Z0[i] = z;
    Zb[0][i] = z;
  }
  __syncthreads();
  if (tid < NTRI) {
    int rem = tid, i = 0, cnt = SHG - 1;
    while (rem >= cnt) { rem -= cnt; ++i; --cnt; }
    int j = i + 1 + rem;
    float v = fmaxf(adj_hidden[g * NTRI + tid], 0.f);
    A[i * SHG + j] = v;
    A[j * SHG + i] = v;
  }
  __syncthreads();

  for (int step = 0; step < NSTEP; ++step) {
    const float* zc = Zb[step & 1];
    float* zn = Zb[(step & 1) ^ 1];
    for (int idx = tid; idx < SHG * HD; idx += 256) {
      int s = idx / HD, h = idx - s * HD;
      float acc = 0.f;
#pragma unroll
      for (int s2 = 0; s2 < SHG; ++s2) acc += A[s * SHG + s2] * zc[s2 * HD + h];
      zn[idx] = acc;
    }
    __syncthreads();
    float* dst = msw + (size_t)(step * HG + g) * (HD * HD);
#pragma unroll
    for (int ii = 0; ii < 16; ++ii) {
      int f = ii * 256 + tid;
      int j = f & 1;
      int lane = (f >> 1) & 31;
      int t = (f >> 6) & 3;
      int k4 = f >> 8;
      int h1 = 4 * k4 + 2 * (lane >> 4) + j;
      int h2 = 16 * t + (lane & 15);
      float acc = 0.f;
#pragma unroll
      for (int s = 0; s < SHG; ++s) acc += Z0[s * HD + h1] * zn[s * HD + h2];
      dst[f] = acc;
    }
    __syncthreads();
  }
}

#ifdef HAVE_TDM
// Issue one TDM descriptor: copy a 4096-element (16 KB) f32 tile from global
// memory into LDS. D# fields per CDNA5 ISA 8.3/8.4 (2D: groups 2/3 unused).
__device__ __forceinline__ void tdm_load_matrix(const float* gsrc, float* ldst) {
  unsigned long long ga = (unsigned long long)(uintptr_t)gsrc;
  unsigned int la = (unsigned int)(uintptr_t)ldst; // low 32b of generic = LDS byte addr
  v4u g0;
  g0[0] = 1u;                                           // count=1 (valid user D#)
  g0[1] = la;                                           // lds_addr
  g0[2] = (unsigned int)ga;                             // global_addr[31:0]
  g0[3] = (unsigned int)((ga >> 32) & 0x1FFFFFFu) | (2u << 30); // addr[56:32] | type=2
  v8i g1;
  g1[0] = 0x00020000;  // workgroup_mask=0, data_size=2 (4 bytes)
  g1[1] = 0x10000000;  // tensor_dim0 = 4096 (low 16 bits live in [31:16])
  g1[2] = 0x00010000;  // tensor_dim0 hi=0; tensor_dim1 = 1
  g1[3] = 0x10000000;  // tensor_dim1 hi=0; tile_dim0 = 4096
  g1[4] = 0x00000001;  // tile_dim1 = 1; tile_dim2 = 0
  g1[5] = 0x00001000;  // tensor_dim0_stride = 4096
  g1[6] = 0;
  g1[7] = 0;
  v4i z4 = {0, 0, 0, 0};
#if __clang_major__ >= 23
  v8i z8 = {0, 0, 0, 0, 0, 0, 0, 0};
  __builtin_amdgcn_tensor_load_to_lds(g0, g1, z4, z4, z8, 0);
#else
  __builtin_amdgcn_tensor_load_to_lds(g0, g1, z4, z4, 0);
#endif
}
#endif

// ---------------------------------------------------------------------------
// Phase 2: out[n, km] = x_n^T M[km] x_n for all 48 matrices.
// 256 threads = 8 waves; each wave owns one 16-row tile of features, held in
// registers in both WMMA A-layout (16 x v2f K-slabs) and C-layout (4 x v8f).
// Matrices are double-buffered in LDS: wave 0 issues a TDM tensor_load_to_lds
// for matrix km+1 while all waves run 64 x v_wmma_f32_16x16x4_f32 on matrix km
// (one barrier per iteration; TENSORcnt hides the 16 KB L2 fetch entirely).
// ---------------------------------------------------------------------------
__global__ __launch_bounds__(256) void rw_main(
    const float* __restrict__ feat,
    const float* __restrict__ msw,
    float* __restrict__ out, int n, int ntiles) {
  __shared__ __align__(16) float Mlds[2][HD * HD];     // 2 x 16 KB double buffer
  __shared__ __align__(16) float red[8 * 32 * 8];      // 8 KB
  __shared__ __align__(16) float outb[8 * 16 * 52];    // 26 KB, pitch 52 vs 48

  const int tid = threadIdx.x;
  const int wave = tid >> 5;
  const int lane = tid & 31;
  const int tIdx = blockIdx.x * 8 + wave;
  const bool active = (tIdx < ntiles); // wave-uniform; EXEC all-ones inside
  const int row0 = tIdx * 16;

#ifdef HAVE_TDM
  if (wave == 0) tdm_load_matrix(msw, &Mlds[0][0]); // prologue prefetch, km=0
#endif

  v2f a[16];
  v8f xc[4];
  if (active) {
    // A-layout: lane l<16 -> row row0+l, cols {4k,4k+1}; l>=16 -> row row0+l-16, cols {4k+2,4k+3}
    const float* pA = feat + (size_t)(row0 + (lane & 15)) * HD + ((lane >> 4) * 2);
#pragma unroll
    for (int k = 0; k < 16; ++k) a[k] = *(const v2f*)(pA + 4 * k);
    // C-layout: reg r, lane l -> X[row0 + r + 8*(l>>4)][16t + (l&15)]
    const float* pC = feat + (size_t)(row0 + 8 * (lane >> 4)) * HD + (lane & 15);
#pragma unroll
    for (int t = 0; t < 4; ++t)
#pragma unroll
      for (int r = 0; r < 8; ++r) xc[t][r] = pC[r * HD + 16 * t];
  }

#pragma unroll 1
  for (int km = 0; km < NMAT; ++km) {
#ifdef HAVE_TDM
    if (wave == 0) __builtin_amdgcn_s_wait_tensorcnt(0); // buf[km&1] landed
    __syncthreads(); // data visible AND everyone done with buf[(km+1)&1]
    if (wave == 0 && km + 1 < NMAT)
      tdm_load_matrix(msw + (size_t)(km + 1) * (HD * HD), &Mlds[(km + 1) & 1][0]);
#else
    { // fallback: synchronous cooperative copy (two barriers)
      const v4f* src = (const v4f*)(msw + (size_t)km * (HD * HD));
      v4f* dstl = (v4f*)&Mlds[km & 1][0];
#pragma unroll
      for (int i = 0; i < 4; ++i) dstl[tid + i * 256] = src[tid + i * 256];
      __syncthreads();
    }
#endif
    if (active) {
      v8f acc[4] = {};
      const v2f* bp = (const v2f*)&Mlds[km & 1][0];
#pragma unroll
      for (int k = 0; k < 16; ++k) {
        v2f b0 = bp[(k * 4 + 0) * 32 + lane];
        v2f b1 = bp[(k * 4 + 1) * 32 + lane];
        v2f b2 = bp[(k * 4 + 2) * 32 + lane];
        v2f b3 = bp[(k * 4 + 3) * 32 + lane];
        acc[0] = __builtin_amdgcn_wmma_f32_16x16x4_f32(
            false, a[k], false, b0, (short)0, acc[0], false, false);
        acc[1] = __builtin_amdgcn_wmma_f32_16x16x4_f32(
            false, a[k], false, b1, (short)0, acc[1], false, false);
        acc[2] = __builtin_amdgcn_wmma_f32_16x16x4_f32(
            false, a[k], false, b2, (short)0, acc[2], false, false);
        acc[3] = __builtin_amdgcn_wmma_f32_16x16x4_f32(
            false, a[k], false, b3, (short)0, acc[3], false, false);
      }
      // per-lane partial of x^T M x: part[r] covers row r+8*(lane>>4), this lane's 4 columns
      v8f part;
#pragma unroll
      for (int r = 0; r < 8; ++r) part[r] = xc[0][r] * acc[0][r];
#pragma unroll
      for (int t = 1; t < 4; ++t)
#pragma unroll
        for (int r = 0; r < 8; ++r) part[r] += xc[t][r] * acc[t][r];

      float* rp = &red[(wave * 32 + lane) * 8];
      *(v4f*)rp = (v4f){part[0], part[1], part[2], part[3]};
      *(v4f*)(rp + 4) = (v4f){part[4], part[5], part[6], part[7]};
      asm volatile("s_wait_dscnt 0x0" ::: "memory"); // cross-lane LDS RAW (same wave)
      if (lane < 16) { // lane R gathers row R across its 16-lane half
        int half = lane >> 3, j = lane & 7;
        const float* gp = &red[(wave * 32 + half * 16) * 8 + j];
        float rs = 0.f;
#pragma unroll
        for (int i2 = 0; i2 < 16; ++i2) rs += gp[i2 * 8];
        outb[wave * (16 * 52) + lane * 52 + km] = rs;
      }
    }
#ifndef HAVE_TDM
    __syncthreads();
#endif
  }

  if (active) { // coalesced b128 flush of 16x48 tile, two identical copies
    asm volatile("s_wait_dscnt 0x0" ::: "memory"); // outb writes (same wave) visible
    const float* ob = &outb[wave * (16 * 52)];
    float* o1 = out + (size_t)row0 * NMAT;
    float* o2 = o1 + (size_t)n * NMAT;
#pragma unroll
    for (int q = 0; q < 6; ++q) {
      int f = q * 128 + lane * 4;
      int row = f / 48, col = f - row * 48;
      v4f v = *(const v4f*)&ob[row * 52 + col];
      *(v4f*)&o1[f] = v;
      *(v4f*)&o2[f] = v;
    }
  }
}

extern "C" void kernel_launch(void* const* d_in, const int* in_sizes, int n_in,
                              void* d_out, int out_size, void* d_ws, size_t ws_size,
                              hipStream_t stream) {
  const float* features = (const float*)d_in[0];
  const float* adj_hidden = (const float*)d_in[1];
  const float* features_hidden = (const float*)d_in[2];
  (void)n_in; (void)out_size; (void)ws_size;

  const int n = in_sizes[0] / HD;      // 200000
  const int ntiles = n / 16;           // 12500 (N divisible by 16)
  float* msw = (float*)d_ws;           // needs NMAT*64*64*4 = 768 KB scratch

  rw_prep<<<HG, 256, 0, stream>>>(adj_hidden, features_hidden, msw);
  const int nblocks = (ntiles + 7) / 8;
  rw_main<<<nblocks, 256, 0, stream>>>(features, msw, (float*)d_out, n, ntiles);
}